// contrative_loss_86225763434863
// MI455X (gfx1250) — compile-verified
//
#include <hip/hip_runtime.h>

typedef float v2f __attribute__((ext_vector_type(2)));
typedef float v8f __attribute__((ext_vector_type(8)));

#define TEMP_INV (1.0f / 0.07f)
#define EPS 1e-5f
#define DDIM 512
#define ROWS_PER_WAVE 16
#define WAVES_PER_BLOCK 8
#define ROWS_PER_BLOCK (ROWS_PER_WAVE * WAVES_PER_BLOCK)   // 128

// One wave (32 lanes) owns a 16-row tile. Using V_WMMA_F32_16X16X4_F32:
//   C1 = X_tile * X_tile^T   -> diag = sim_self
//   C2 = X_tile * G_tile^T   -> diag = sim_neg   (G = gathered rows x[s0[.]])
// A-fragment layout (ISA 7.12.2, 32-bit A 16x4): lane L holds row (L&15),
// K = k + 2*(L>>4) .. +1  -> one float2 per lane per 4-wide K step.
// For B = (rows)^T the per-lane data is identical, so the same float2 feeds
// both the A and B operands.
__global__ __launch_bounds__(256)
void contrastive_loss_main(const float* __restrict__ x,
                           const int* __restrict__ y,
                           const int* __restrict__ samples,  // (N,2) int32 pairs
                           float* __restrict__ partial)
{
    const int lane = threadIdx.x & 31;
    const int wave = threadIdx.x >> 5;
    const int rb   = blockIdx.x * ROWS_PER_BLOCK + wave * ROWS_PER_WAVE;

    const int j = lane & 15;   // tile row this lane streams
    const int h = lane >> 4;   // half-wave -> K sub-offset 2*h

    const size_t myrow = (size_t)(rb + j);
    const int    g0    = samples[2 * myrow];          // s0 index for row rb+j
    const float* xrow  = x + myrow * DDIM + 2 * h;
    const float* grow  = x + (size_t)g0 * DDIM + 2 * h;

    v8f c1 = {};  // X * X^T accumulator (diag = sim_self)
    v8f c2 = {};  // X * G^T accumulator (diag = sim_neg)

#pragma unroll 8
    for (int k = 0; k < DDIM; k += 4) {
        v2f a = *(const v2f*)(xrow + k);
        v2f b = *(const v2f*)(grow + k);
        // 8 args: (neg_a, A, neg_b, B, c_mod, C, reuse_a, reuse_b)
        c1 = __builtin_amdgcn_wmma_f32_16x16x4_f32(false, a, false, a,
                                                   (short)0, c1, false, false);
        c2 = __builtin_amdgcn_wmma_f32_16x16x4_f32(false, a, false, b,
                                                   (short)0, c2, false, false);
    }

    // Diagonal ownership: C layout = VGPR r, half h -> M = r + 8h, N = lane&15.
    // Diagonal element i sits in lane 16*(i>>3)+i, register i&7.
    // => lanes 0..7 own rows rb+0..7 (c[j]); lanes 24..31 own rows rb+8..15 (c[j-8]).
    const bool owns = (h == 0) ? (j < 8) : (j >= 8);
    const int  r    = (h == 0) ? j : (j - 8);

    float sim_self = 0.0f, sim_neg = 0.0f;
#pragma unroll
    for (int rr = 0; rr < 8; ++rr) {
        if (rr == r) { sim_self = c1[rr]; sim_neg = c2[rr]; }
    }

    float loss = 0.0f;
    if (owns) {
        const size_t row = (size_t)(rb + j);      // owned diagonal index == j
        const int s0r = samples[2 * row];
        const int s1r = samples[2 * row + 1];
        const float label = (y[s0r] != y[s1r]) ? 1.0f : 0.0f;
        const float diff  = (sim_self - sim_neg) * TEMP_INV;
        const float p     = 1.0f / (1.0f + expf(-diff));
        loss = -(logf(p + EPS) * label + logf(1.0f - p + EPS) * (1.0f - label));
    }

    // Deterministic wave32 tree reduction.
    float v = loss;
#pragma unroll
    for (int off = 16; off >= 1; off >>= 1)
        v += __shfl_xor(v, off, 32);

    __shared__ float wsum[WAVES_PER_BLOCK];
    if (lane == 0) wsum[wave] = v;
    __syncthreads();
    if (threadIdx.x == 0) {
        float s = 0.0f;
#pragma unroll
        for (int w = 0; w < WAVES_PER_BLOCK; ++w) s += wsum[w];
        partial[blockIdx.x] = s;
    }
}

// Single-block deterministic tree reduction of per-block partials; writes mean.
__global__ __launch_bounds__(256)
void contrastive_loss_reduce(const float* __restrict__ partial,
                             float* __restrict__ out,
                             int nblocks, float invN)
{
    __shared__ float sm[256];
    float s = 0.0f;
    for (int i = threadIdx.x; i < nblocks; i += 256) s += partial[i];
    sm[threadIdx.x] = s;
    __syncthreads();
#pragma unroll
    for (int st = 128; st > 0; st >>= 1) {
        if ((int)threadIdx.x < st) sm[threadIdx.x] += sm[threadIdx.x + st];
        __syncthreads();
    }
    if (threadIdx.x == 0) out[0] = sm[0] * invN;
}

extern "C" void kernel_launch(void* const* d_in, const int* in_sizes, int n_in,
                              void* d_out, int out_size, void* d_ws, size_t ws_size,
                              hipStream_t stream) {
    const float* x       = (const float*)d_in[0];
    const int*   y       = (const int*)d_in[1];
    const int*   samples = (const int*)d_in[2];   // int32 pairs (JAX demotes int64)

    const int N       = in_sizes[1];              // y has N elements
    const int nblocks = N / ROWS_PER_BLOCK;       // 262144/128 = 2048
    float* partial    = (float*)d_ws;             // 2048 * 4B = 8 KB scratch

    contrastive_loss_main<<<nblocks, 256, 0, stream>>>(x, y, samples, partial);
    contrastive_loss_reduce<<<1, 256, 0, stream>>>(partial, (float*)d_out,
                                                   nblocks, 1.0f / (float)N);
}